// NoisyTopkRouter_7911329759613
// MI455X (gfx1250) — compile-verified
//
#include <hip/hip_runtime.h>
#include <hip/hip_bf16.h>
#include <float.h>

typedef float v2f __attribute__((ext_vector_type(2)));
typedef float v8f __attribute__((ext_vector_type(8)));

#define BB 4
#define NN 4096
#define DD 2048
#define EE 128
#define KK 8

#define ROWS 64          // rows per workgroup
#define KC   32          // K chunk staged in LDS
#define XST  36          // LDS stride (floats) for x tile (mult of 4 -> 16B alignment)
#define WST  36          // LDS stride (floats) for W tile
#define LST  129         // LDS stride for logits tile

#define XS_FLOATS    (ROWS * XST)             // 2304
#define WS_FLOATS    (EE * WST)               // 4608
#define CHUNK_FLOATS (XS_FLOATS + WS_FLOATS)  // 6912 per buffer
#define SMEM_FLOATS  (2 * CHUNK_FLOATS)       // 13824 floats = 55,296 B (logits needs 8256)

// CDNA5 async copy: 16 B from global memory into LDS; tracked by ASYNCcnt.
__device__ __forceinline__ void async_copy_b128(const float* gsrc, const float* lds_dst) {
    // Generic LDS address: addr[31:0] is the workgroup-relative LDS byte offset.
    unsigned int       loff  = (unsigned int)(size_t)lds_dst;
    unsigned long long gaddr = (unsigned long long)(size_t)gsrc;
    asm volatile("global_load_async_to_lds_b128 %0, %1, off"
                 :: "v"(loff), "v"(gaddr) : "memory");
}
__device__ __forceinline__ void wait_async0() {
    asm volatile("s_wait_asynccnt 0x0" ::: "memory");
}

__global__ __launch_bounds__(256) void moe_router_kernel(
    const float* __restrict__ x, const float* __restrict__ W,
    const float* __restrict__ bias, float* __restrict__ out)
{
    __shared__ float smem[SMEM_FLOATS];
    float* lg = smem;            // [64][129] logits tile (reused after GEMM)

    const int tid     = threadIdx.x;
    const int lane    = tid & 31;
    const int wave    = tid >> 5;
    const int rowBase = blockIdx.x * ROWS;

    const int rgrp  = (wave & 3) * 16;    // wave's 16-row group within the 64-row tile
    const int eb    = (wave >> 2) * 64;   // wave's expert base (4 tiles of 16 experts)
    const int l15   = lane & 15;
    const int khalf = (lane >> 4) * 2;    // A/B frag: lanes 16-31 hold K+2,K+3

    v8f acc0 = {}, acc1 = {}, acc2 = {}, acc3 = {};

    // issue async stage of one K-chunk into buffer p (6 x b128 per thread)
    auto stage = [&](int p, int kb) {
        float* xs = smem + p * CHUNK_FLOATS;
        float* ws = xs + XS_FLOATS;
        int q = tid;
        #pragma unroll
        for (int i = 0; i < 2; ++i) {          // x tile: 64 x 32 floats
            int r  = q >> 3;
            int c4 = (q & 7) * 4;
            async_copy_b128(x + (size_t)(rowBase + r) * DD + kb + c4,
                            xs + r * XST + c4);
            q += 256;
        }
        q = tid;
        #pragma unroll
        for (int i = 0; i < 4; ++i) {          // W tile: 128 x 32 floats
            int r  = q >> 3;
            int c4 = (q & 7) * 4;
            async_copy_b128(W + (size_t)r * DD + kb + c4,
                            ws + r * WST + c4);
            q += 256;
        }
    };

    // prologue: chunk 0 -> buffer 0
    stage(0, 0);
    wait_async0();
    __syncthreads();

    int p = 0;
    for (int kb = 0; kb < DD; kb += KC) {
        if (kb + KC < DD) stage(p ^ 1, kb + KC);   // overlap next chunk with compute

        const float* xs = smem + p * CHUNK_FLOATS;
        const float* ws = xs + XS_FLOATS;
        #pragma unroll
        for (int s = 0; s < KC / 4; ++s) {
            int kl = s * 4 + khalf;
            v2f a  = *(const v2f*)(xs + (rgrp + l15) * XST + kl);
            v2f b0 = *(const v2f*)(ws + (eb +  0 + l15) * WST + kl);
            v2f b1 = *(const v2f*)(ws + (eb + 16 + l15) * WST + kl);
            v2f b2 = *(const v2f*)(ws + (eb + 32 + l15) * WST + kl);
            v2f b3 = *(const v2f*)(ws + (eb + 48 + l15) * WST + kl);
            acc0 = __builtin_amdgcn_wmma_f32_16x16x4_f32(false, a, false, b0, (short)0, acc0, false, false);
            acc1 = __builtin_amdgcn_wmma_f32_16x16x4_f32(false, a, false, b1, (short)0, acc1, false, false);
            acc2 = __builtin_amdgcn_wmma_f32_16x16x4_f32(false, a, false, b2, (short)0, acc2, false, false);
            acc3 = __builtin_amdgcn_wmma_f32_16x16x4_f32(false, a, false, b3, (short)0, acc3, false, false);
        }

        wait_async0();      // my async loads into the other buffer are done
        __syncthreads();    // everyone finished compute on buf p and their loads
        p ^= 1;
    }

    __syncthreads();
    // C layout: VGPR j = row (j + 8*(lane>>4)), lane&15 = col. Write (+bias) into LDS logits tile.
    {
        int n     = l15;
        int mhalf = (lane >> 4) * 8;
        #pragma unroll
        for (int t = 0; t < 4; ++t) {
            int col  = eb + t * 16 + n;
            float bv = bias[col];
            v8f acc  = (t == 0) ? acc0 : (t == 1) ? acc1 : (t == 2) ? acc2 : acc3;
            #pragma unroll
            for (int j = 0; j < 8; ++j) {
                lg[(rgrp + mhalf + j) * LST + col] = acc[j] + bv;
            }
        }
    }
    __syncthreads();

    // ---- per-row top-8 (stable: ties keep lowest index) + softmax ----
    float tv[KK];
    int   ti[KK];
    float gates[KK];
    const int r = tid;
    if (tid < ROWS) {
        #pragma unroll
        for (int k = 0; k < KK; ++k) { tv[k] = -FLT_MAX; ti[k] = 0; }
        for (int e = 0; e < EE; ++e) {
            float g = lg[r * LST + e];
            if (g > tv[KK - 1]) {            // strict > : incumbent (lower idx) wins ties
                tv[KK - 1] = g; ti[KK - 1] = e;
                #pragma unroll
                for (int j = KK - 1; j > 0; --j) {
                    if (tv[j] > tv[j - 1]) { // strict > : stable insertion
                        float tf = tv[j]; tv[j] = tv[j - 1]; tv[j - 1] = tf;
                        int   tt = ti[j]; ti[j] = ti[j - 1]; ti[j - 1] = tt;
                    }
                }
            }
        }
        float mx = tv[0];
        float s  = 0.f;
        #pragma unroll
        for (int k = 0; k < KK; ++k) { gates[k] = __expf(tv[k] - mx); s += gates[k]; }
        float inv = 1.f / s;
        #pragma unroll
        for (int k = 0; k < KK; ++k) gates[k] *= inv;
    }
    __syncthreads();

    // zero dense tile in LDS
    for (int q = tid; q < ROWS * EE; q += 256)
        lg[(q >> 7) * LST + (q & 127)] = 0.f;
    __syncthreads();

    // scatter gates into LDS; write indices (as float) to global
    if (tid < ROWS) {
        float* idxOut = out + (size_t)BB * NN * EE;
        #pragma unroll
        for (int k = 0; k < KK; ++k) {
            lg[r * LST + ti[k]] = gates[k];
            idxOut[(size_t)(rowBase + r) * KK + k] = (float)ti[k];
        }
    }
    __syncthreads();

    // coalesced writeout of dense gates
    for (int q = tid; q < ROWS * EE; q += 256) {
        int rr = q >> 7, cc = q & 127;
        out[(size_t)(rowBase + rr) * EE + cc] = lg[rr * LST + cc];
    }
}

extern "C" void kernel_launch(void* const* d_in, const int* in_sizes, int n_in,
                              void* d_out, int out_size, void* d_ws, size_t ws_size,
                              hipStream_t stream) {
    (void)in_sizes; (void)n_in; (void)out_size; (void)d_ws; (void)ws_size;
    const float* x = (const float*)d_in[0];
    const float* W = (const float*)d_in[1];
    const float* b = (const float*)d_in[2];
    float* out = (float*)d_out;
    const int totalRows = BB * NN;              // 16384
    dim3 grid(totalRows / ROWS);                // 256 blocks
    moe_router_kernel<<<grid, 256, 0, stream>>>(x, W, b, out);
}